// SparseResidualMoE_81999515615502
// MI455X (gfx1250) — compile-verified
//
#include <hip/hip_runtime.h>
#include <hip/hip_bf16.h>
#include <math.h>
#include <stdint.h>

typedef __attribute__((ext_vector_type(2))) float v2f;
typedef __attribute__((ext_vector_type(8))) float v8f;
typedef __attribute__((ext_vector_type(4))) float f4v;

#define DIM   128
#define NTGT  4096
#define MREF  65536
#define KTOP  32
#define EPSW  1e-6f

// ---------------- workspace layout (float offsets) ----------------
#define WS_SUM   0
#define WS_SSQ   128
#define WS_MEAN  256
#define WS_ISTD  384
#define WS_SQR   512
#define WS_SQT   (512 + 65536)
#define WS_STGT  (WS_SQT + 4096)
#define WS_SREF  (WS_STGT + 4096*128)
// total floats = WS_SREF + 65536*128  (~36 MB)

// ---------------- output layout (float offsets, return order) ------
#define OFF_PROTO  0                       // 4096*128
#define OFF_DELTA  (OFF_PROTO + 524288)    // 4096*128
#define OFF_ABSD   (OFF_DELTA + 524288)    // 4096*128
#define OFF_TOP1   (OFF_ABSD  + 524288)    // 4096
#define OFF_MEAND  (OFF_TOP1  + 4096)      // 4096
#define OFF_STDD   (OFF_MEAND + 4096)      // 4096
#define OFF_TIDX   (OFF_STDD  + 4096)      // 4096*32 (int32 bits)
#define OFF_TDIST  (OFF_TIDX  + 131072)    // 4096*32
#define OFF_PW     (OFF_TDIST + 131072)    // 4096*32
#define OFF_SAME   (OFF_PW    + 131072)    // 4096

// B tile in LDS: 128 refs x 128 dims, padded row stride 132 floats
// (132 % 64 = 4 -> 16 lanes at stride 4 banks: conflict-free; 528B row
//  stride keeps 16B alignment for B128 async stores)
#define BSTR  132
#define TILEF (128 * BSTR)   // floats per B buffer

__global__ void init_ws_kernel(float* ws) {
    if (threadIdx.x < 256) ws[threadIdx.x] = 0.0f;  // sum + sumsq accumulators
}

__global__ void stats_kernel(const float* __restrict__ ref, float* __restrict__ ws) {
    const int c    = threadIdx.x & 127;
    const int rh   = threadIdx.x >> 7;
    const int base = blockIdx.x * 256 + rh * 128;
    float s = 0.0f, q = 0.0f;
    for (int i = 0; i < 128; ++i) {
        float v = ref[(size_t)(base + i) * DIM + c];
        s += v; q += v * v;
    }
    atomicAdd(&ws[WS_SUM + c], s);
    atomicAdd(&ws[WS_SSQ + c], q);
}

__global__ void finalize_stats_kernel(float* __restrict__ ws) {
    const int c = threadIdx.x;  // 128 threads
    float mean = ws[WS_SUM + c] * (1.0f / 65536.0f);
    float var  = ws[WS_SSQ + c] * (1.0f / 65536.0f) - mean * mean;
    var = fmaxf(var, 0.0f);
    float sd = sqrtf(var);
    ws[WS_MEAN + c] = mean;
    ws[WS_ISTD + c] = (sd > 0.0f) ? (1.0f / sd) : 1.0f;
}

// scale rows by (x-mean)*istd, emit scaled rows + row squared norms.
__global__ void scale_rows_kernel(const float* __restrict__ src, int nrows,
                                  const float* __restrict__ ws,
                                  float* __restrict__ outv,
                                  float* __restrict__ outsq) {
    const float* mean = ws + WS_MEAN;
    const float* istd = ws + WS_ISTD;
    const int lane = threadIdx.x & 31;
    const int wave = threadIdx.x >> 5;
    const int base = blockIdx.x * 64;
    for (int rl = 0; rl < 8; ++rl) {
        int row = base + rl * 8 + wave;           // uniform per wave
        if (row >= nrows) continue;
        float s = 0.0f;
#pragma unroll
        for (int cc = 0; cc < 4; ++cc) {
            int c = lane + cc * 32;
            float v = (src[(size_t)row * DIM + c] - mean[c]) * istd[c];
            outv[(size_t)row * DIM + c] = v;
            s += v * v;
        }
        for (int off = 16; off > 0; off >>= 1) s += __shfl_xor(s, off, 32);
        if (lane == 0) outsq[row] = s;
    }
}

// Fused: fp32 WMMA distance GEMM (double-buffered LDS fed by CDNA5 async
// global->LDS copies, grouped B-fragment prefetch) + streaming top-32 +
// full epilogue.  grid = 256 blocks (16 targets each), 256 threads (8 waves).
__global__ void topk_proto_kernel(const float* __restrict__ tgt,
                                  const float* __restrict__ ref,
                                  const int*   __restrict__ tdom,
                                  const int*   __restrict__ rdom,
                                  const float* __restrict__ ws,
                                  float* __restrict__ out) {
    extern __shared__ char smem_raw[];
    // NOTE: no static __shared__ in this kernel -> dynamic LDS starts at
    // offset 0, so byte offsets into this layout are LDS addresses for the
    // async-to-LDS instructions.
    float* Bl     = (float*)smem_raw;            // 2 x 128*132 (double buffer)
    float* sqrl   = Bl + 2 * TILEF;              // 2 x 128
    float* sqtl   = sqrl + 256;                  // 16
    float* dtile  = sqtl + 16;                   // 16*128
    float* topd   = dtile + 16 * 128;            // 16*32
    int*   topi   = (int*)(topd + 16 * 32);      // 16*32
    float* qd     = (float*)(topi + 16 * 32);    // 16*128 (reused as pw later)
    int*   qi     = (int*)(qd + 16 * 128);       // 16*128
    float* thresh = (float*)(qi + 16 * 128);     // 16
    int*   qn     = (int*)(thresh + 16);         // 16

    const float* stgt = ws + WS_STGT;
    const float* sref = ws + WS_SREF;
    const float* sqt  = ws + WS_SQT;
    const float* sqr  = ws + WS_SQR;

    const int tid  = threadIdx.x;
    const int lane = tid & 31;
    const int wave = tid >> 5;
    const int tb   = blockIdx.x * 16;            // target base

    if (tid < 16) { thresh[tid] = 3.0e38f; qn[tid] = 0; sqtl[tid] = sqt[tb + tid]; }
    for (int i = tid; i < 16 * 32; i += 256) { topd[i] = 3.0e38f; topi[i] = 0; }

    // ---- A fragments held in registers for the whole scan ----
    // fp32 16x16x4 A layout: m = lane%16; VGPR0 K = khalf, VGPR1 K = khalf+1
    const int m     = lane & 15;
    const int khalf = (lane >> 4) << 1;          // 0 or 2
    v2f afrag[32];
#pragma unroll
    for (int ks = 0; ks < 32; ++ks) {
        const float* ap = stgt + (size_t)(tb + m) * DIM + ks * 4 + khalf;
        afrag[ks].x = ap[0];
        afrag[ks].y = ap[1];
    }

    const int nloc = wave * 16 + (lane & 15);    // this lane's ref column in 128-tile
    const int moff = (lane >> 4) << 3;           // 0 or 8 (C/D row offset)
    const int r0   = tid >> 5;                   // staging: base row handled by thread
    const int c40  = tid & 31;                   // staging: float4 column

    // ---- prologue: stage tile 0 into buffer 0 (plain loads, one-time) ----
    {
        const f4v* src4 = (const f4v*)sref;
#pragma unroll
        for (int u = 0; u < 16; ++u) {
            int r = r0 + u * 8;
            *(f4v*)(Bl + r * BSTR + c40 * 4) = src4[r * 32 + c40];
        }
        if (tid < 128) sqrl[tid] = sqr[tid];
    }
    __syncthreads();

    int cur = 0;
    for (int tile = 0; tile < MREF; tile += 128) {
        const float* Bc  = Bl + cur * TILEF;
        const float* sqc = sqrl + cur * 128;
        const int    nxt = cur ^ 1;
        const bool   havenext = (tile + 128) < MREF;

        // ---- kick off async global->LDS copy of the next B tile ----
        // GLOBAL_LOAD_ASYNC_TO_LDS_B128, GVS mode: SGPR base (uniform next
        // tile) + per-lane 32-bit byte offset; LDS dest address in VGPR.
        // No VGPRs hold data; tracked by ASYNCcnt, overlaps the WMMA chain
        // and the filter/insert phases below.
        float sqnext = 0.0f;
        if (havenext) {
            const float* gbase = sref + (size_t)(tile + 128) * DIM;
            uint32_t ldsoff = (uint32_t)(nxt * TILEF * 4 + r0 * (BSTR * 4) + c40 * 16);
            uint32_t goff   = (uint32_t)(r0 * 512 + c40 * 16);
#pragma unroll
            for (int u = 0; u < 16; ++u) {
                asm volatile("global_load_async_to_lds_b128 %0, %1, %2"
                             :
                             : "v"(ldsoff), "v"(goff), "s"(gbase)
                             : "memory");
                ldsoff += 8 * BSTR * 4;   // 8 rows of padded LDS tile
                goff   += 8 * 512;        // 8 rows of global tile
            }
            if (tid < 128) sqnext = sqr[tile + 128 + tid];
        }

        // ---- 16x16 GEMM per wave, K=128: 4 groups of (8 ds_load -> 8 wmma) ----
        v8f c0 = {}, c1 = {};
        const float* bbase = Bc + nloc * BSTR + khalf;
#pragma unroll
        for (int g = 0; g < 4; ++g) {
            v2f bf[8];
#pragma unroll
            for (int u = 0; u < 8; ++u) {
                int ks = g * 8 + u;
                bf[u].x = bbase[ks * 4 + 0];
                bf[u].y = bbase[ks * 4 + 1];
            }
#pragma unroll
            for (int u = 0; u < 8; ++u) {
                int ks = g * 8 + u;
                if (u & 1)
                    c1 = __builtin_amdgcn_wmma_f32_16x16x4_f32(
                            false, afrag[ks], false, bf[u], (short)0, c1, false, false);
                else
                    c0 = __builtin_amdgcn_wmma_f32_16x16x4_f32(
                            false, afrag[ks], false, bf[u], (short)0, c0, false, false);
            }
        }

        // d = sqrt(max(||t||^2 + ||r||^2 - 2*dot, 0))
#pragma unroll
        for (int i = 0; i < 8; ++i) {
            int mm = i + moff;
            float dot = c0[i] + c1[i];
            float d2  = sqtl[mm] + sqc[nloc] - 2.0f * dot;
            dtile[mm * 128 + nloc] = sqrtf(fmaxf(d2, 0.0f));
        }
        __syncthreads();   // dtile visible

        // ---- parallel filter: 16 threads per target vs current threshold ----
        {
            int   t  = tid >> 4;
            int   j0 = tid & 15;
            float th = thresh[t];
            for (int s = 0; s < 8; ++s) {
                int   j = j0 + s * 16;
                float d = dtile[t * 128 + j];
                if (d < th) {
                    int slot = atomicAdd(&qn[t], 1);
                    qd[t * 128 + slot] = d;
                    qi[t * 128 + slot] = tile + j;
                }
            }
        }
        __syncthreads();

        // ---- serial sorted insertion, one thread per target ----
        if (tid < 16) {
            int    t  = tid;
            int    nq = qn[t];
            float* td = topd + t * 32;
            int*   ti = topi + t * 32;
            for (int q = 0; q < nq; ++q) {
                float d = qd[t * 128 + q];
                if (d < td[31]) {
                    int idx = qi[t * 128 + q];
                    int pos = 31;
                    while (pos > 0 && td[pos - 1] > d) {
                        td[pos] = td[pos - 1]; ti[pos] = ti[pos - 1]; --pos;
                    }
                    td[pos] = d; ti[pos] = idx;
                }
            }
            thresh[t] = td[31];
            qn[t] = 0;
        }

        // ---- drain async copies, publish next buffer + sqr row norms ----
        if (havenext) {
            if (tid < 128) sqrl[nxt * 128 + tid] = sqnext;
            asm volatile("s_wait_asynccnt 0x0" ::: "memory");
        }
        __syncthreads();   // thresh/qn + Bl[nxt] + sqrl[nxt] visible

        cur = nxt;
    }

    // ---------------- epilogue ----------------
    float* pwl = qd;  // reuse 16x128 region for proto weights (32 used per row)
    if (tid < 16) {
        int t = tid;
        const float* td = topd + t * 32;
        const int*   ti = topi + t * 32;
        float s = 0.0f;
        float rw[32];
#pragma unroll
        for (int k = 0; k < 32; ++k) { rw[k] = 1.0f / fmaxf(td[k], EPSW); s += rw[k]; }
        float inv = 1.0f / s;
        float meand = 0.0f;
#pragma unroll
        for (int k = 0; k < 32; ++k) {
            float pw = rw[k] * inv;
            pwl[t * 128 + k] = pw;
            meand += pw * td[k];
        }
        float var = 0.0f;
        int   same = 0;
        int   mydom = tdom[tb + t];
        for (int k = 0; k < 32; ++k) {
            float dd = td[k] - meand;
            var += pwl[t * 128 + k] * dd * dd;
            same += (rdom[ti[k]] == mydom) ? 1 : 0;
        }
        out[OFF_TOP1  + tb + t] = td[0];
        out[OFF_MEAND + tb + t] = meand;
        out[OFF_STDD  + tb + t] = sqrtf(fmaxf(var, 0.0f));
        out[OFF_SAME  + tb + t] = (float)same * (1.0f / 32.0f);
    }
    __syncthreads();

    for (int i = tid; i < 16 * 32; i += 256) {
        int t = i >> 5, k = i & 31;
        ((int*)(out + OFF_TIDX))[(tb + t) * 32 + k] = topi[t * 32 + k];
        out[OFF_TDIST + (size_t)(tb + t) * 32 + k] = topd[t * 32 + k];
        out[OFF_PW    + (size_t)(tb + t) * 32 + k] = pwl[t * 128 + k];
    }

    // prototypes / delta / |delta|: 2 targets x 128 dims per pass
    for (int p = 0; p < 8; ++p) {
        int t  = p * 2 + (tid >> 7);
        int dc = tid & 127;
        float acc = 0.0f;
#pragma unroll 4
        for (int k = 0; k < 32; ++k)
            acc += pwl[t * 128 + k] * ref[(size_t)topi[t * 32 + k] * DIM + dc];
        float tv = tgt[(size_t)(tb + t) * DIM + dc];
        float dl = tv - acc;
        out[OFF_PROTO + (size_t)(tb + t) * DIM + dc] = acc;
        out[OFF_DELTA + (size_t)(tb + t) * DIM + dc] = dl;
        out[OFF_ABSD  + (size_t)(tb + t) * DIM + dc] = fabsf(dl);
    }
}

extern "C" void kernel_launch(void* const* d_in, const int* in_sizes, int n_in,
                              void* d_out, int out_size, void* d_ws, size_t ws_size,
                              hipStream_t stream) {
    const float* tgt  = (const float*)d_in[0];   // 4096x128
    const float* ref  = (const float*)d_in[1];   // 65536x128
    const int*   tdom = (const int*)d_in[2];     // 4096
    const int*   rdom = (const int*)d_in[3];     // 65536
    float*       out  = (float*)d_out;
    float*       ws   = (float*)d_ws;

    (void)in_sizes; (void)n_in; (void)out_size; (void)ws_size;

    init_ws_kernel<<<1, 256, 0, stream>>>(ws);
    stats_kernel<<<256, 256, 0, stream>>>(ref, ws);
    finalize_stats_kernel<<<1, 128, 0, stream>>>(ws);
    scale_rows_kernel<<<MREF / 64, 256, 0, stream>>>(ref, MREF, ws,
                                                     ws + WS_SREF, ws + WS_SQR);
    scale_rows_kernel<<<NTGT / 64, 256, 0, stream>>>(tgt, NTGT, ws,
                                                     ws + WS_STGT, ws + WS_SQT);

    size_t smem = (size_t)(2 * TILEF + 256 + 16 + 16 * 128 + 16 * 32) * 4  // Bl x2, sqrl x2, sqtl, dtile, topd
                + (size_t)(16 * 32) * 4                                     // topi
                + (size_t)(16 * 128) * 4 * 2                                // qd, qi
                + 16 * 4 * 2;                                               // thresh, qn
    topk_proto_kernel<<<NTGT / 16, 256, smem, stream>>>(tgt, ref, tdom, rdom, ws, out);
}